// VGAE_31671088840936
// MI455X (gfx1250) — compile-verified
//
#include <hip/hip_runtime.h>

typedef __attribute__((ext_vector_type(2))) float v2f;
typedef __attribute__((ext_vector_type(8))) float v8f;

#define N_NODES 8192
#define E_EDGES 262144
#define DIM_IN  512
#define H1      32
#define H2      16
#define EPS_VAL 1e-32f

// ---------------------------------------------------------------- zero fill
__global__ void vgae_zero(float* __restrict__ p, int n) {
    int i = blockIdx.x * blockDim.x + threadIdx.x;
    if (i < n) p[i] = 0.0f;
}

// ---------------------------------------------------------------- hw fp32 atomic add
// Non-returning global_atomic_add_f32 at device scope (accumulation crosses
// workgroups; atomic units live in L2). Tracked by STOREcnt; implicit
// s_wait_idle before s_endpgm guarantees completion at kernel end.
__device__ __forceinline__ void hw_atomic_add_f32(float* addr, float v) {
    asm volatile("global_atomic_add_f32 %0, %1, off scope:SCOPE_DEV"
                 :
                 : "v"(addr), "v"(v)
                 : "memory");
}

// ---------------------------------------------------------------- fp32 WMMA GEMM
// out[M x Nc] = X[M x K] @ W[K x Nc]; one wave per 16x16 tile.
// V_WMMA_F32_16X16X4_F32 layout:
//   A(16x4): lane = (k/2)*16 + m, vgpr = k%2   -> float2 per lane, contiguous in k
//   B(4x16): lane = (k/2)*16 + n, vgpr = k%2
//   C/D(16x16): vgpr v -> M = v + 8*(lane/16), N = lane%16
__global__ void vgae_gemm_wmma(const float* __restrict__ X,
                               const float* __restrict__ W,
                               float* __restrict__ out,
                               int M, int K, int Nc) {
    int wave   = (blockIdx.x * blockDim.x + threadIdx.x) >> 5;
    int tilesN = Nc >> 4;
    int ti = wave / tilesN;
    int tj = wave % tilesN;
    int lane = threadIdx.x & 31;
    int m  = lane & 15;
    int kh = lane >> 4;            // which K-pair (0: k=0,1  1: k=2,3) within a slice

    const float* arow = X + (size_t)(ti * 16 + m) * K + kh * 2;
    const float* bcol = W + (size_t)(kh * 2) * Nc + tj * 16 + m;

    v8f c = {};
    for (int kk = 0; kk < K; kk += 4) {
        v2f a = *(const v2f*)(arow + kk);
        v2f b;
        b.x = bcol[(size_t)kk * Nc];
        b.y = bcol[(size_t)(kk + 1) * Nc];
        c = __builtin_amdgcn_wmma_f32_16x16x4_f32(false, a, false, b,
                                                  (short)0, c, false, false);
    }
#pragma unroll
    for (int v = 0; v < 8; ++v) {
        out[(size_t)(ti * 16 + kh * 8 + v) * Nc + tj * 16 + m] = c[v];
    }
}

// ---------------------------------------------------------------- COO SpMM scatter
// one lane per (edge, dim); D is a power of two (32 or 16).
// row/col/val broadcast within the wave; feature-row accesses are contiguous.
template <int D>
__global__ void vgae_spmm_scatter(const int* __restrict__ row,
                                  const int* __restrict__ col,
                                  const float* __restrict__ val,
                                  const float* __restrict__ x,
                                  float* __restrict__ out) {
    unsigned idx = blockIdx.x * blockDim.x + threadIdx.x;
    unsigned e = idx / D;
    unsigned d = idx % D;
    int r  = row[e];
    int cc = col[e];
    float v = val[e];
    hw_atomic_add_f32(&out[(size_t)r * D + d], v * x[(size_t)cc * D + d]);
}

// ---------------------------------------------------------------- relu(h) @ W1
// one thread per output element of [N, H2]
__global__ void vgae_relu_gemm(const float* __restrict__ h,
                               const float* __restrict__ W1,
                               float* __restrict__ out) {
    int idx = blockIdx.x * blockDim.x + threadIdx.x;  // N*H2 threads
    int i = idx >> 4;
    int j = idx & 15;
    float s = 0.0f;
#pragma unroll
    for (int k = 0; k < H1; ++k)
        s = fmaf(fmaxf(h[(size_t)i * H1 + k], 0.0f), W1[k * H2 + j], s);
    out[idx] = s;
}

// ---------------------------------------------------------------- mean = max(mean, eps)
__global__ void vgae_eps(const float* __restrict__ in, float* __restrict__ out, int n) {
    int i = blockIdx.x * blockDim.x + threadIdx.x;
    if (i < n) out[i] = fmaxf(in[i], EPS_VAL);
}

// ---------------------------------------------------------------- decode: sigmoid(Z Z^T)
// one wave per 16x16 tile of the 8192x8192 output; K = 16 -> 4 chained WMMAs.
// B(k,n) = Z[n,k] so per-lane addressing for the B slice is identical to A
// with the column-tile base (contiguous float2 from a row of Z).
// Output is a 256MB write-once stream (> 192MB L2): store non-temporal so the
// hot 512KB Z stays L2-resident and the stream goes straight to HBM.
__global__ void vgae_decode_wmma(const float* __restrict__ Z,
                                 float* __restrict__ out) {
    int wave = (blockIdx.x * blockDim.x + threadIdx.x) >> 5;
    int ti = wave >> 9;        // / 512 tiles per row
    int tj = wave & 511;
    int lane = threadIdx.x & 31;
    int m  = lane & 15;
    int kh = lane >> 4;

    const float* arow = Z + (size_t)(ti * 16 + m) * H2 + kh * 2;
    const float* brow = Z + (size_t)(tj * 16 + m) * H2 + kh * 2;

    v8f c = {};
#pragma unroll
    for (int kk = 0; kk < H2; kk += 4) {
        v2f a = *(const v2f*)(arow + kk);
        v2f b = *(const v2f*)(brow + kk);
        c = __builtin_amdgcn_wmma_f32_16x16x4_f32(false, a, false, b,
                                                  (short)0, c, false, false);
    }
#pragma unroll
    for (int v = 0; v < 8; ++v) {
        float x = c[v];
        float s = 1.0f / (1.0f + __expf(-x));
        __builtin_nontemporal_store(
            s, &out[(size_t)(ti * 16 + kh * 8 + v) * N_NODES + tj * 16 + m]);
    }
}

// ---------------------------------------------------------------- launcher
extern "C" void kernel_launch(void* const* d_in, const int* in_sizes, int n_in,
                              void* d_out, int out_size, void* d_ws, size_t ws_size,
                              hipStream_t stream) {
    const float* X        = (const float*)d_in[0];
    const int*   edge_row = (const int*)  d_in[1];
    const int*   edge_col = (const int*)  d_in[2];
    const float* edge_val = (const float*)d_in[3];
    const float* W0       = (const float*)d_in[4];
    const float* W1       = (const float*)d_in[5];
    // d_in[6] = W2: std path never reaches the output -> skipped

    float* out = (float*)d_out;

    // workspace layout (floats)
    float* ws   = (float*)d_ws;
    float* xw0  = ws;                         // N * H1 = 262144
    float* h    = xw0 + (size_t)N_NODES * H1; // N * H1 = 262144
    float* hw1  = h   + (size_t)N_NODES * H1; // N * H2 = 131072
    float* zraw = hw1 + (size_t)N_NODES * H2; // N * H2 = 131072
    float* z    = zraw + (size_t)N_NODES * H2;// N * H2 = 131072

    const int TB = 256;

    // zero the scatter accumulators (ws is poisoned; must be per-call deterministic)
    vgae_zero<<<(N_NODES * H1) / TB, TB, 0, stream>>>(h, N_NODES * H1);
    vgae_zero<<<(N_NODES * H2) / TB, TB, 0, stream>>>(zraw, N_NODES * H2);

    // XW0 = X @ W0 : tiles = (8192/16)*(32/16) = 1024 waves -> 128 blocks
    {
        int waves  = (N_NODES / 16) * (H1 / 16);
        int blocks = waves * 32 / TB;
        vgae_gemm_wmma<<<blocks, TB, 0, stream>>>(X, W0, xw0, N_NODES, DIM_IN, H1);
    }

    // h = spmm(adj, XW0)
    vgae_spmm_scatter<H1><<<(E_EDGES * H1) / TB, TB, 0, stream>>>(
        edge_row, edge_col, edge_val, xw0, h);

    // hw1 = relu(h) @ W1
    vgae_relu_gemm<<<(N_NODES * H2) / TB, TB, 0, stream>>>(h, W1, hw1);

    // zraw = spmm(adj, hw1)
    vgae_spmm_scatter<H2><<<(E_EDGES * H2) / TB, TB, 0, stream>>>(
        edge_row, edge_col, edge_val, hw1, zraw);

    // z = max(zraw, eps)
    vgae_eps<<<(N_NODES * H2) / TB, TB, 0, stream>>>(zraw, z, N_NODES * H2);

    // out = sigmoid(z @ z^T): 512*512 tiles -> 262144 waves -> 32768 blocks
    {
        int waves  = (N_NODES / 16) * (N_NODES / 16);
        int blocks = waves / (TB / 32);
        vgae_decode_wmma<<<blocks, TB, 0, stream>>>(z, out);
    }
}